// MoELayer_20590073217781
// MI455X (gfx1250) — compile-verified
//
#include <hip/hip_runtime.h>

// Problem constants (from reference)
#define D_ 256
#define E_ 1024
#define F_ 1024
#define NROWS_ 4096

typedef __attribute__((ext_vector_type(2))) float v2f;
typedef __attribute__((ext_vector_type(8))) float v8f;

// ---------------------------------------------------------------------------
// Shared fp32 WMMA GEMM-with-bias:  C[M x 1024] = A[M x 256] * B[256 x 1024] + bias
// One wave computes one 16x16 output tile via 64 chained V_WMMA_F32_16X16X4_F32.
// A-layout per lane (16x4 f32): m = lane&15, holds K = 2*(lane>>4) + {0,1}.
// B-layout per lane (4x16 f32): n = lane&15, vgpr j holds K = 2*(lane>>4) + j.
// C/D layout: vgpr r -> row (r + 8*(lane>>4)), col = lane&15.
// ---------------------------------------------------------------------------
__global__ __launch_bounds__(256)
void gemm_bias_wmma_f32(const float* __restrict__ A,
                        const float* __restrict__ B,
                        const float* __restrict__ bias,
                        float* __restrict__ C,
                        int tilesN) {
    const int lane = threadIdx.x & 31;
    const int wave = threadIdx.x >> 5;
    const int tile = blockIdx.x * 8 + wave;
    const int tm   = (tile / tilesN) * 16;
    const int tn   = (tile % tilesN) * 16;
    const int half = lane >> 4;     // 0 or 1
    const int mr   = lane & 15;

    const float* arow = A + (size_t)(tm + mr) * D_;
    const float* bcol = B + (tn + mr);

    v8f acc = {};
#pragma unroll 4
    for (int k0 = 0; k0 < D_; k0 += 4) {
        const int ka = k0 + 2 * half;
        v2f a = *(const v2f*)(arow + ka);          // two consecutive K, b64 load
        v2f b;
        b.x = bcol[(size_t)(ka + 0) * 1024];
        b.y = bcol[(size_t)(ka + 1) * 1024];
        acc = __builtin_amdgcn_wmma_f32_16x16x4_f32(
            /*neg_a=*/false, a, /*neg_b=*/false, b,
            /*c_mod=*/(short)0, acc, /*reuse_a=*/false, /*reuse_b=*/false);
    }

    const int   col = tn + mr;
    const float bv  = bias[col];
#pragma unroll
    for (int r = 0; r < 8; ++r) {
        C[(size_t)(tm + r + 8 * half) * 1024 + col] = acc[r] + bv;
    }
}

// ---------------------------------------------------------------------------
// Row softmax in place: one block per row of 1024, 256 threads x float4.
// ---------------------------------------------------------------------------
__global__ __launch_bounds__(256)
void softmax_rows(float* __restrict__ w) {
    __shared__ float red[256];
    const int t = threadIdx.x;
    float4* row = (float4*)(w + (size_t)blockIdx.x * E_);

    float4 v = row[t];
    float m = fmaxf(fmaxf(v.x, v.y), fmaxf(v.z, v.w));
    red[t] = m;
    __syncthreads();
#pragma unroll
    for (int s = 128; s > 0; s >>= 1) {
        if (t < s) red[t] = fmaxf(red[t], red[t + s]);
        __syncthreads();
    }
    m = red[0];
    __syncthreads();

    v.x = __expf(v.x - m);
    v.y = __expf(v.y - m);
    v.z = __expf(v.z - m);
    v.w = __expf(v.w - m);
    red[t] = v.x + v.y + v.z + v.w;
    __syncthreads();
#pragma unroll
    for (int s = 128; s > 0; s >>= 1) {
        if (t < s) red[t] += red[t + s];
        __syncthreads();
    }
    const float inv = 1.0f / red[0];
    v.x *= inv; v.y *= inv; v.z *= inv; v.w *= inv;
    row[t] = v;
}

// ---------------------------------------------------------------------------
// Expert mix (the HBM-bound part: streams 1.07 GB of expert_W exactly once):
//   W_eff[d,f] = sum_e w[e,f] * expert_W[e,d,f]
//   b_eff[f]   = sum_e w[e,f] * expert_b[e,f]   (computed by block d==0)
// Block = one d (256 blocks), thread = 4 consecutive f (float4 / b128 loads).
// w columns are revisited by all 256 blocks -> served from L2 (4 MB << 192 MB).
// ---------------------------------------------------------------------------
__global__ __launch_bounds__(256)
void mix_experts(const float* __restrict__ w,
                 const float* __restrict__ eW,
                 const float* __restrict__ eb,
                 float* __restrict__ Weff,
                 float* __restrict__ beff) {
    const int d  = blockIdx.x;          // 0..255
    const int f0 = threadIdx.x * 4;     // 0..1020
    const bool do_bias = (d == 0);

    float4 acc  = {0.f, 0.f, 0.f, 0.f};
    float4 accb = {0.f, 0.f, 0.f, 0.f};

    for (int e = 0; e < E_; ++e) {
        const float4 wv = *(const float4*)(w  + (size_t)e * F_ + f0);
        const float4 ev = *(const float4*)(eW + ((size_t)e * D_ + d) * F_ + f0);
        if (e + 4 < E_) {  // stay ahead of the 1 GB stream
            __builtin_prefetch(eW + ((size_t)(e + 4) * D_ + d) * F_ + f0, 0, 1);
        }
        acc.x += wv.x * ev.x;
        acc.y += wv.y * ev.y;
        acc.z += wv.z * ev.z;
        acc.w += wv.w * ev.w;
        if (do_bias) {
            const float4 bv = *(const float4*)(eb + (size_t)e * F_ + f0);
            accb.x += wv.x * bv.x;
            accb.y += wv.y * bv.y;
            accb.z += wv.z * bv.z;
            accb.w += wv.w * bv.w;
        }
    }
    *(float4*)(Weff + (size_t)d * F_ + f0) = acc;
    if (do_bias) {
        *(float4*)(beff + f0) = accb;
    }
}

// ---------------------------------------------------------------------------
extern "C" void kernel_launch(void* const* d_in, const int* in_sizes, int n_in,
                              void* d_out, int out_size, void* d_ws, size_t ws_size,
                              hipStream_t stream) {
    const float* x  = (const float*)d_in[0];   // (N, D)
    const float* gW = (const float*)d_in[1];   // (D, E)
    const float* gb = (const float*)d_in[2];   // (E,)
    const float* eW = (const float*)d_in[3];   // (E, D, F)
    const float* eb = (const float*)d_in[4];   // (E, F)
    float* out = (float*)d_out;                // (N, F)

    float* ws   = (float*)d_ws;
    float* w    = ws;                           // (E, E)  = 4 MB
    float* Weff = ws + (size_t)E_ * E_;         // (D, F)  = 1 MB
    float* beff = Weff + (size_t)D_ * F_;       // (F,)    = 4 KB

    // 1) Gate logits for first E rows only: w = x[:E] @ gate_W + gate_b
    //    tiles = (E/16)*(E/16) = 4096; 8 waves/block -> 512 blocks
    gemm_bias_wmma_f32<<<512, 256, 0, stream>>>(x, gW, gb, w, E_ / 16);

    // 2) Row softmax in place
    softmax_rows<<<E_, 256, 0, stream>>>(w);

    // 3) Expert mix: W_eff, b_eff (HBM-bound: 1.07 GB streamed once)
    mix_experts<<<D_, 256, 0, stream>>>(w, eW, eb, Weff, beff);

    // 4) out = x @ W_eff + b_eff
    //    tiles = (N/16)*(F/16) = 16384; 8 waves/block -> 2048 blocks
    gemm_bias_wmma_f32<<<2048, 256, 0, stream>>>(x, Weff, beff, out, F_ / 16);
}